// Dimension_63101659513158
// MI455X (gfx1250) — compile-verified
//
#include <hip/hip_runtime.h>

typedef __attribute__((ext_vector_type(2))) float v2f;
typedef __attribute__((ext_vector_type(4))) float v4f;
typedef __attribute__((ext_vector_type(8))) float v8f;

#define N_PTS       8192
#define DIMS        32
#define KSEL        16              // k (top-k list size, includes self)
#define ROWS_PER_WG 128             // 8 waves x 16 rows
#define WAVES       8
#define COLS_PER_CHUNK 32           // two 16x16 C tiles -> two independent WMMA chains
#define CHUNKS      (N_PTS / COLS_PER_CHUNK)   // 256
#define TILE_PITCH  36              // floats; 144B rows: 16B-aligned, 36*lo mod 64 is a permutation

// ---------------------------------------------------------------------------
// Main kernel: per-batch Gram matrix via V_WMMA_F32_16X16X4_F32 (two
// interleaved accumulation chains for ILP), streaming top-16 selection per
// row, deterministic per-workgroup partial sum of s_i.
// ---------------------------------------------------------------------------
__global__ __launch_bounds__(256) void mle_topk_kernel(const float* __restrict__ X,
                                                       float* __restrict__ partials,
                                                       int nWgPerBatch)
{
    __shared__ __attribute__((aligned(16))) float sqnorm[N_PTS];               // 32 KB
    __shared__ __attribute__((aligned(16))) float tile[WAVES][16][TILE_PITCH]; // 18 KB; reused as merge scratch
    __shared__ float wsum[WAVES];

    const int wg   = blockIdx.x;
    const int b    = wg / nWgPerBatch;
    const int rblk = wg % nWgPerBatch;
    const float* __restrict__ Xb = X + (size_t)b * N_PTS * DIMS;

    const int tid  = threadIdx.x;
    const int lane = tid & 31;
    const int wave = tid >> 5;
    const int lo   = lane & 15;   // row (A) / col (B) index within 16
    const int hi   = lane >> 4;   // K-group select for A/B frags, row-group for C

    // ---- Phase 0: row squared norms into LDS (batch is L2-resident) ----
    for (int i = tid; i < N_PTS; i += 256) {
        const float4* p = (const float4*)(Xb + (size_t)i * DIMS);
        float acc = 0.f;
#pragma unroll
        for (int q = 0; q < DIMS / 4; ++q) {
            float4 v = p[q];
            acc += v.x * v.x + v.y * v.y + v.z * v.z + v.w * v.w;
        }
        sqnorm[i] = acc;
    }
    __syncthreads();

    // ---- Phase 1: load A fragments once (16 rows x 32 dims, fp32) ----
    // V_WMMA_F32_16X16X4_F32 A layout: lane m / m+16 hold row m;
    // VGPR0/1 = K {0,1} (lanes 0-15) or K {2,3} (lanes 16-31).
    const int r0 = rblk * ROWS_PER_WG + wave * 16;
    v2f afrag[8];
#pragma unroll
    for (int kk = 0; kk < 8; ++kk) {
        const float* p = Xb + (size_t)(r0 + lo) * DIMS + kk * 4 + hi * 2;
        afrag[kk].x = p[0];
        afrag[kk].y = p[1];
    }

    // Row norms for this lane's C rows (m = r + 8*hi), invariant over chunks.
    float sqi[8];
#pragma unroll
    for (int r = 0; r < 8; ++r) sqi[r] = sqnorm[r0 + r + 8 * hi];

    // Per-lane sorted ascending top-16 list (lane owns half of row `lo`).
    float t[KSEL];
#pragma unroll
    for (int q = 0; q < KSEL; ++q) t[q] = 3.0e38f;

    float (*mytile)[TILE_PITCH] = tile[wave];

    // ---- Phase 2: sweep all 8192 columns in 32-wide chunks ----
    for (int c = 0; c < CHUNKS; ++c) {
        const int j0 = c * COLS_PER_CHUNK;
        if (c + 1 < CHUNKS) {  // warm next chunk's B rows (gfx1250 global_prefetch)
            __builtin_prefetch(Xb + (size_t)(j0 + 32 + lo) * DIMS, 0, 0);
            __builtin_prefetch(Xb + (size_t)(j0 + 48 + lo) * DIMS, 0, 0);
        }

        // Two independent WMMA accumulation chains (cols j0.. and j0+16..):
        // breaks the serial D->C dependency so the matrix pipe stays fed.
        v8f acc0 = {};
        v8f acc1 = {};
#pragma unroll
        for (int kk = 0; kk < 8; ++kk) {
            const float* p0 = Xb + (size_t)(j0 + lo) * DIMS + kk * 4 + hi * 2;
            const float* p1 = p0 + 16 * DIMS;
            v2f b0; b0.x = p0[0]; b0.y = p0[1];
            v2f b1; b1.x = p1[0]; b1.y = p1[1];
            acc0 = __builtin_amdgcn_wmma_f32_16x16x4_f32(
                false, afrag[kk], false, b0, (short)0, acc0, false, false);
            acc1 = __builtin_amdgcn_wmma_f32_16x16x4_f32(
                false, afrag[kk], false, b1, (short)0, acc1, false, false);
        }

        // d2 = |xi|^2 + |xj|^2 - 2*<xi,xj>; scatter both tiles into LDS.
        const float sqj0 = sqnorm[j0 + lo];
        const float sqj1 = sqnorm[j0 + 16 + lo];
#pragma unroll
        for (int r = 0; r < 8; ++r) {
            mytile[r + 8 * hi][lo]      = sqi[r] + sqj0 - 2.0f * acc0[r];
            mytile[r + 8 * hi][16 + lo] = sqi[r] + sqj1 - 2.0f * acc1[r];
        }

        // Intra-wave RAW through LDS: DS ops are in-order per wave; make the
        // dependency explicit with the CDNA5 split wait counter.
        asm volatile("s_wait_dscnt 0" ::: "memory");

        // Each lane scans 16 values of its row-half (four aligned b128 loads).
        const v4f* rp = (const v4f*)&mytile[lo][hi * 16];
        v4f f0 = rp[0];
        v4f f1 = rp[1];
        v4f f2 = rp[2];
        v4f f3 = rp[3];
        float vals[16] = {f0.x, f0.y, f0.z, f0.w, f1.x, f1.y, f1.z, f1.w,
                          f2.x, f2.y, f2.z, f2.w, f3.x, f3.y, f3.z, f3.w};
#pragma unroll
        for (int q = 0; q < 16; ++q) {
            float v = vals[q];
            if (v < t[KSEL - 1]) {          // accepted ~k*ln(n/k) times total
                t[KSEL - 1] = v;
#pragma unroll
                for (int s = KSEL - 1; s >= 1; --s) {   // branchless bubble insert
                    float a_ = fminf(t[s - 1], t[s]);
                    float b_ = fmaxf(t[s - 1], t[s]);
                    t[s - 1] = a_;
                    t[s]     = b_;
                }
            }
        }
    }

    // ---- Phase 3: merge the two half-row lists (partner = lane ^ 16) ----
    float o[KSEL];
#pragma unroll
    for (int q = 0; q < KSEL; ++q) o[q] = __shfl_xor(t[q], 16, 32);

    float s_i = 0.0f;
    if (hi == 0) {
        // Tile rows are dead now; reuse row `lo` (36 >= 32 floats) as merge buf.
        float* mb = &mytile[lo][0];
#pragma unroll
        for (int q = 0; q < KSEL; ++q) { mb[q] = t[q]; mb[KSEL + q] = o[q]; }
        asm volatile("s_wait_dscnt 0" ::: "memory");

        // Two-pointer merge of two sorted 16-lists; position 0 is self.
        // s_i = 0.5*( 15*L15 - sum_{q=1..15} L_q ),  L = log(max(d2, 1e-12)).
        int ia = 0, ib = 0;
        float lsum = 0.0f, last = 0.0f;
        for (int q = 0; q < KSEL; ++q) {
            float va = mb[ia];
            float vb = mb[KSEL + ib];
            float v;
            if (va <= vb) { v = va; ++ia; } else { v = vb; ++ib; }
            float lg = __logf(fmaxf(v, 1e-12f));
            if (q >= 1) lsum += lg;
            last = lg;
        }
        s_i = 0.5f * (15.0f * last - lsum);
    }

    // ---- Phase 4: deterministic reduction (shfl tree -> LDS -> one lane) ----
#pragma unroll
    for (int m = 8; m >= 1; m >>= 1) s_i += __shfl_xor(s_i, m, 32);
    if (lane == 0) wsum[wave] = s_i;
    __syncthreads();
    if (tid == 0) {
        float tot = 0.0f;
#pragma unroll
        for (int w = 0; w < WAVES; ++w) tot += wsum[w];
        partials[wg] = tot;
    }
}

// ---------------------------------------------------------------------------
// Finalize: fixed-order sum of per-WG partials; dim = (k-2)*N / sum(s_i).
// ---------------------------------------------------------------------------
__global__ void mle_finalize_kernel(const float* __restrict__ partials,
                                    const int* __restrict__ kptr,
                                    float* __restrict__ out,
                                    int B, int nWgPerBatch)
{
    int b = blockIdx.x * blockDim.x + threadIdx.x;
    if (b >= B) return;
    float S = 0.0f;
    for (int i = 0; i < nWgPerBatch; ++i) S += partials[b * nWgPerBatch + i];
    int k = kptr ? kptr[0] : KSEL;
    float Km1 = (float)(k - 2);   // (K-1) with K = k-1
    out[b] = Km1 * (float)N_PTS / S;
}

extern "C" void kernel_launch(void* const* d_in, const int* in_sizes, int n_in,
                              void* d_out, int out_size, void* d_ws, size_t ws_size,
                              hipStream_t stream)
{
    const float* X    = (const float*)d_in[0];
    const int*   kptr = (n_in > 1) ? (const int*)d_in[1] : nullptr;
    float*       out  = (float*)d_out;
    float*       partials = (float*)d_ws;   // B * (N/ROWS_PER_WG) floats, fully overwritten

    const int B   = in_sizes[0] / (N_PTS * DIMS);   // 4
    const int nWg = N_PTS / ROWS_PER_WG;            // 64

    mle_topk_kernel<<<dim3(B * nWg), dim3(256), 0, stream>>>(X, partials, nWg);
    mle_finalize_kernel<<<dim3(1), dim3(64), 0, stream>>>(partials, kptr, out, B, nWg);
}